// SimpleMambaLM_17635135717648
// MI455X (gfx1250) — compile-verified
//
#include <hip/hip_runtime.h>
#include <hip/hip_bf16.h>
#include <math.h>

// ---- problem constants (from reference) ----
constexpr int Vc   = 32000;
constexpr int DMc  = 1024;
constexpr int DSc  = 16;
constexpr int NLc  = 2;
constexpr int DCc  = 4;
constexpr int DIc  = 2048;   // EXP * DM
constexpr int DTRc = 64;     // DM / 16
constexpr int Bc   = 2;
constexpr int Lc   = 1024;
constexpr int XPW  = DTRc + 2 * DSc;   // 96
constexpr int ROWS = Bc * Lc;          // 2048 token rows

typedef __attribute__((ext_vector_type(2))) float        v2f;
typedef __attribute__((ext_vector_type(8))) float        v8f;
typedef __attribute__((ext_vector_type(4))) unsigned int v4u;
typedef __attribute__((ext_vector_type(4))) int          v4i;
typedef __attribute__((ext_vector_type(8))) int          v8i;

// A-tile LDS row stride (16 data floats + 2 pad floats from TDM pad feature):
// keeps float2 fragment loads 8B-aligned and the 16-row gather conflict-free.
constexpr int APAD = 18;

__device__ __forceinline__ unsigned lds_off_of(const void* p) {
    // generic pointer to LDS: addr[31:0] is the LDS byte offset (ISA §10.2)
    return (unsigned)(unsigned long long)(uintptr_t)p;
}

// ---------------------------------------------------------------------------
// Issue one 2D TDM tile load: tileY rows x tileX f32 elems, row stride
// 'strideElems' (f32 units), into LDS at ldsOff.  'ctrl' = D# group1 word0
// (data_size / pad config).  Wave-level op, tracked with TENSORcnt.
// D# packing per CDNA5 ISA.
// ---------------------------------------------------------------------------
__device__ __forceinline__ void tdm_load_tile(const float* g, unsigned ldsOff,
                                              unsigned tileX, unsigned tileY,
                                              unsigned long long strideElems,
                                              unsigned ctrl)
{
    const unsigned long long ga = (unsigned long long)(uintptr_t)g;

    v4u g0;
    g0.x = 1u;                                        // count=1, user D#
    g0.y = ldsOff;                                    // lds_addr [63:32]
    g0.z = (unsigned)(ga & 0xffffffffu);              // global_addr lo
    g0.w = (unsigned)((ga >> 32) & 0x01ffffffu)       // global_addr hi (57b)
         | (2u << 30);                                // type = 2 ("image")

    v8i g1;
    g1[0] = (int)ctrl;                                // mask=0, data_size, pad cfg
    g1[1] = (int)((tileX & 0xffffu) << 16);           // tensor_dim0[15:0] (=tile)
    g1[2] = (int)(((tileX >> 16) & 0xffffu) | ((tileY & 0xffffu) << 16));
    g1[3] = (int)(((tileY >> 16) & 0xffffu) | ((tileX & 0xffffu) << 16)); // tile_dim0
    g1[4] = (int)(tileY & 0xffffu);                   // tile_dim1 (tile_dim2=0)
    g1[5] = (int)(strideElems & 0xffffffffu);         // tensor_dim0_stride lo
    g1[6] = (int)((strideElems >> 32) & 0xffffu);     // stride hi; dim1_stride=0
    g1[7] = 0;

    const v4i z4 = {0, 0, 0, 0};
#if defined(__clang_major__) && (__clang_major__ >= 23)
    const v8i z8 = {0, 0, 0, 0, 0, 0, 0, 0};
    __builtin_amdgcn_tensor_load_to_lds(g0, g1, z4, z4, z8, 0);
#else
    __builtin_amdgcn_tensor_load_to_lds(g0, g1, z4, z4, 0);
#endif
}

// group1 word0 values: data_size=4B (bits 17:16 = 2)
constexpr unsigned TDM_CTRL_PLAIN = (2u << 16);
// + pad_enable(b20) + pad_interval=3 (16 DWORDs, b24:22) + pad_amount=1 (2 DWORDs, b31:25)
constexpr unsigned TDM_CTRL_PAD18 = (2u << 16) | (1u << 20) | (3u << 22) | (1u << 25);

// ---------------------------------------------------------------------------
// TDM-fed fp32 WMMA GEMM: C[M,N] = A[M,K] * B[K,N], M%64==0, N%64==0, K%16==0.
// 128 threads = 4 waves per block; block computes a 64x64 C tile.
// Double-buffered LDS tiles (A: 64x16 padded to stride 18, B: 16x64) filled by
// tensor_load_to_lds issued from wave 0; s_wait_tensorcnt + barrier per chunk;
// next chunk's DMA issued before computing the current one (overlap).
// ---------------------------------------------------------------------------
__global__ __launch_bounds__(128) void gemm_wmma_tdm(
    const float* __restrict__ A, int lda,
    const float* __restrict__ Bm, int ldb,
    float* __restrict__ C, int ldc, int K)
{
    __shared__ float As[2][64 * APAD];
    __shared__ float Bs[2][16 * 64];

    const int tid  = threadIdx.x;
    const int w    = tid >> 5;     // wave 0..3 -> rows 16w..16w+15 of the tile
    const int lane = tid & 31;
    const int half = lane >> 4;    // K-pair select per ISA f32 A/B layout
    const int l16  = lane & 15;
    const int bn   = blockIdx.x * 64;
    const int bm   = blockIdx.y * 64;

    const int nchunks = K >> 4;

    if (tid < 32) {   // TDM ignores EXEC -> keep it behind a wave-level branch
        tdm_load_tile(A + (size_t)bm * lda, lds_off_of(&As[0][0]),
                      16, 64, (unsigned long long)lda, TDM_CTRL_PAD18);
        tdm_load_tile(Bm + (size_t)bn, lds_off_of(&Bs[0][0]),
                      64, 16, (unsigned long long)ldb, TDM_CTRL_PLAIN);
    }

    v8f acc[4] = {};
    __builtin_amdgcn_s_wait_tensorcnt(0);   // no-op for non-issuing waves
    __syncthreads();

    for (int c = 0; c < nchunks; ++c) {
        const int cur = c & 1;
        const int nxt = cur ^ 1;

        if ((c + 1 < nchunks) && tid < 32) {    // prefetch next chunk via TDM
            const int k = (c + 1) << 4;
            tdm_load_tile(A + (size_t)bm * lda + k, lds_off_of(&As[nxt][0]),
                          16, 64, (unsigned long long)lda, TDM_CTRL_PAD18);
            tdm_load_tile(Bm + (size_t)k * ldb + bn, lds_off_of(&Bs[nxt][0]),
                          64, 16, (unsigned long long)ldb, TDM_CTRL_PLAIN);
        }

#pragma unroll
        for (int k4 = 0; k4 < 4; ++k4) {
            // A fragment: lanes 0-15 carry K={4k4,4k4+1}, lanes 16-31 K={+2,+3}
            const float2 af = *(const float2*)
                &As[cur][(16 * w + l16) * APAD + k4 * 4 + half * 2];
            v2f a; a.x = af.x; a.y = af.y;
#pragma unroll
            for (int j = 0; j < 4; ++j) {       // 4 N-tiles reuse the A frag
                v2f b;
                b.x = Bs[cur][(k4 * 4 + half * 2    ) * 64 + j * 16 + l16];
                b.y = Bs[cur][(k4 * 4 + half * 2 + 1) * 64 + j * 16 + l16];
                acc[j] = __builtin_amdgcn_wmma_f32_16x16x4_f32(
                    false, a, false, b, (short)0, acc[j], false, false);
            }
        }

        __builtin_amdgcn_s_wait_tensorcnt(0);
        __syncthreads();
    }

    // C/D layout: VGPR r -> row half*8 + r, col = l16
    float* crow = C + (size_t)(bm + 16 * w + half * 8) * ldc + bn + l16;
#pragma unroll
    for (int j = 0; j < 4; ++j)
#pragma unroll
        for (int r = 0; r < 8; ++r)
            crow[(size_t)r * ldc + j * 16] = acc[j][r];
}

// ---------------------------------------------------------------------------
// Simple one-wave WMMA GEMM (kept for the narrow N=96 x-proj GEMM).
// ---------------------------------------------------------------------------
__global__ __launch_bounds__(32) void gemm_wmma_f32(
    const float* __restrict__ A, int lda,
    const float* __restrict__ Bm, int ldb,
    float* __restrict__ C, int ldc,
    int K)
{
    const int lane = threadIdx.x;
    const int half = lane >> 4;
    const int l16  = lane & 15;
    const int tn   = blockIdx.x * 16;
    const int tm   = blockIdx.y * 16;

    v8f acc = {};
    const float* arow = A  + (size_t)(tm + l16) * lda + half * 2;
    const float* bcol = Bm + (size_t)(half * 2) * ldb + tn + l16;

    for (int k = 0; k < K; k += 4) {
        const float2 av = *(const float2*)(arow + k);
        v2f a; a.x = av.x; a.y = av.y;
        v2f b;
        b.x = bcol[(size_t)k       * ldb];
        b.y = bcol[(size_t)(k + 1) * ldb];
        acc = __builtin_amdgcn_wmma_f32_16x16x4_f32(
            false, a, false, b, (short)0, acc, false, false);
    }

    float* crow = C + (size_t)(tm + half * 8) * ldc + tn + l16;
#pragma unroll
    for (int r = 0; r < 8; ++r)
        crow[(size_t)r * ldc] = acc[r];
}

// ---------------------------------------------------------------------------
// Embedding gather: one block per token row, 256 threads x float4 = 1024 f32
// ---------------------------------------------------------------------------
__global__ __launch_bounds__(256) void embed_gather(
    const int* __restrict__ ids, const float* __restrict__ emb,
    float* __restrict__ x)
{
    const int row = blockIdx.x;
    const int idx = ids[row];
    const float4* src = (const float4*)(emb + (size_t)idx * DMc);
    float4*       dst = (float4*)(x + (size_t)row * DMc);
    dst[threadIdx.x] = src[threadIdx.x];
}

// ---------------------------------------------------------------------------
// Depthwise causal conv (DC=4 taps) + bias + SiLU.  Reads xin = xz[:, :DI].
// ---------------------------------------------------------------------------
__global__ __launch_bounds__(256) void conv_silu(
    const float* __restrict__ xz, const float* __restrict__ cw,
    const float* __restrict__ cb, float* __restrict__ u)
{
    const int i = blockIdx.x * blockDim.x + threadIdx.x;
    const int d = i % DIc;
    const int l = (i / DIc) % Lc;
    const int b = i / (DIc * Lc);

    float acc = cb[d];
#pragma unroll
    for (int k = 0; k < DCc; ++k) {
        const int ls = l - (DCc - 1) + k;
        if (ls >= 0)
            acc += xz[(size_t)(b * Lc + ls) * (2 * DIc) + d] * cw[d * DCc + k];
    }
    const float sig = 1.f / (1.f + __expf(-acc));
    u[i] = acc * sig;
}

// ---------------------------------------------------------------------------
// delta = softplus(dtWdt + bias), numerically stable, in place.
// ---------------------------------------------------------------------------
__global__ __launch_bounds__(256) void softplus_bias(
    float* __restrict__ delta, const float* __restrict__ bdt)
{
    const int i = blockIdx.x * blockDim.x + threadIdx.x;
    const float t = delta[i] + bdt[i % DIc];
    delta[i] = fmaxf(t, 0.f) + log1pf(__expf(-fabsf(t)));
}

// ---------------------------------------------------------------------------
// Fused selective scan.  16 lanes = one (b,d) channel; lane = state index n.
// Keeps h in registers (avoids ~0.5 GB dA/dBu tensors), reduces over n with
// shfl_xor inside 16-lane halves of the wave32, and fuses +u*D and the
// SiLU(z) gate into the writeout.
// ---------------------------------------------------------------------------
__global__ __launch_bounds__(256) void mamba_scan(
    const float* __restrict__ delta, const float* __restrict__ u,
    const float* __restrict__ dbc,   const float* __restrict__ xz,
    const float* __restrict__ A_log, const float* __restrict__ Dv,
    float* __restrict__ y)
{
    const int t = blockIdx.x * blockDim.x + threadIdx.x;
    const int n = t & (DSc - 1);
    const int g = t >> 4;            // b*DI + d
    const int d = g & (DIc - 1);
    const int b = g >> 11;           // DIc == 2^11

    const float Aneg = -__expf(A_log[d * DSc + n]);
    const float Dval = Dv[d];

    float h = 0.f;
    for (int l = 0; l < Lc; ++l) {
        const size_t row = (size_t)(b * Lc + l);
        const float dl = delta[row * DIc + d];
        const float uv = u[row * DIc + d];
        const float Bv = dbc[row * XPW + DTRc + n];
        const float Cv = dbc[row * XPW + DTRc + DSc + n];

        h = __expf(dl * Aneg) * h + (dl * uv) * Bv;

        float p = h * Cv;
        p += __shfl_xor(p, 1, 32);
        p += __shfl_xor(p, 2, 32);
        p += __shfl_xor(p, 4, 32);
        p += __shfl_xor(p, 8, 32);

        if (n == 0) {
            const float zv  = xz[row * (2 * DIc) + DIc + d];
            const float sig = 1.f / (1.f + __expf(-zv));
            y[row * DIc + d] = (p + uv * Dval) * (zv * sig);
        }
    }
}

// ---------------------------------------------------------------------------
// LayerNorm (population variance), in place, one block per row.
// ---------------------------------------------------------------------------
__global__ __launch_bounds__(256) void layernorm_inplace(
    float* __restrict__ x, const float* __restrict__ g,
    const float* __restrict__ bt)
{
    __shared__ float ssum[256];
    __shared__ float ssq[256];
    float* xr = x + (size_t)blockIdx.x * DMc;

    float s = 0.f, sq = 0.f;
    for (int i = threadIdx.x; i < DMc; i += 256) {
        const float v = xr[i];
        s += v; sq += v * v;
    }
    ssum[threadIdx.x] = s; ssq[threadIdx.x] = sq;
    __syncthreads();
    for (int st = 128; st > 0; st >>= 1) {
        if (threadIdx.x < st) {
            ssum[threadIdx.x] += ssum[threadIdx.x + st];
            ssq[threadIdx.x]  += ssq[threadIdx.x + st];
        }
        __syncthreads();
    }
    const float mu   = ssum[0] * (1.f / DMc);
    const float var  = ssq[0] * (1.f / DMc) - mu * mu;
    const float rstd = rsqrtf(var + 1e-5f);
    for (int i = threadIdx.x; i < DMc; i += 256)
        xr[i] = (xr[i] - mu) * rstd * g[i] + bt[i];
}

// ---------------------------------------------------------------------------
// Host launch
// ---------------------------------------------------------------------------
extern "C" void kernel_launch(void* const* d_in, const int* in_sizes, int n_in,
                              void* d_out, int out_size, void* d_ws, size_t ws_size,
                              hipStream_t stream)
{
    const int*   ids     = (const int*)  d_in[0];
    const float* emb     = (const float*)d_in[1];
    const float* in_proj = (const float*)d_in[2];
    const float* conv_w  = (const float*)d_in[3];
    const float* conv_b  = (const float*)d_in[4];
    const float* x_proj  = (const float*)d_in[5];
    const float* dt_w    = (const float*)d_in[6];
    const float* dt_b    = (const float*)d_in[7];
    const float* A_log   = (const float*)d_in[8];
    const float* Dmat    = (const float*)d_in[9];
    const float* out_w   = (const float*)d_in[10];
    const float* ln_g    = (const float*)d_in[11];
    const float* ln_b    = (const float*)d_in[12];
    const float* head    = (const float*)d_in[13];
    float*       out     = (float*)d_out;

    // workspace layout (floats)
    float* ws = (float*)d_ws;
    const size_t SZ_X   = (size_t)ROWS * DMc;
    const size_t SZ_XZ  = (size_t)ROWS * 2 * DIc;
    const size_t SZ_U   = (size_t)ROWS * DIc;
    const size_t SZ_DBC = (size_t)ROWS * XPW;
    float* x     = ws;
    float* xz    = x     + SZ_X;
    float* u     = xz    + SZ_XZ;
    float* dbc   = u     + SZ_U;
    float* delta = dbc   + SZ_DBC;
    float* yb    = delta + SZ_U;

    // 1) embedding
    embed_gather<<<ROWS, 256, 0, stream>>>(ids, emb, x);

    for (int layer = 0; layer < NLc; ++layer) {
        const float* Wi  = in_proj + (size_t)layer * DMc * 2 * DIc;
        const float* cw  = conv_w  + (size_t)layer * DIc * DCc;
        const float* cb  = conv_b  + (size_t)layer * DIc;
        const float* Wx  = x_proj  + (size_t)layer * DIc * XPW;
        const float* Wdt = dt_w    + (size_t)layer * DTRc * DIc;
        const float* bdt = dt_b    + (size_t)layer * DIc;
        const float* Al  = A_log   + (size_t)layer * DIc * DSc;
        const float* Dl  = Dmat    + (size_t)layer * DIc;
        const float* Wo  = out_w   + (size_t)layer * DIc * DMc;

        // xz = x @ Wi   (2048 x 1024 x 4096)  [TDM + WMMA]
        gemm_wmma_tdm<<<dim3(2 * DIc / 64, ROWS / 64), 128, 0, stream>>>(
            x, DMc, Wi, 2 * DIc, xz, 2 * DIc, DMc);

        // u = silu(conv(xin) + cb)
        conv_silu<<<(ROWS * DIc) / 256, 256, 0, stream>>>(xz, cw, cb, u);

        // dbc = u @ Wx  (2048 x 2048 x 96)  [N=96: simple WMMA path]
        gemm_wmma_f32<<<dim3(XPW / 16, ROWS / 16), 32, 0, stream>>>(
            u, DIc, Wx, XPW, dbc, XPW, DIc);

        // delta_pre = dt @ Wdt : A = dbc[:, :64] with lda = 96  [TDM + WMMA]
        gemm_wmma_tdm<<<dim3(DIc / 64, ROWS / 64), 128, 0, stream>>>(
            dbc, XPW, Wdt, DIc, delta, DIc, DTRc);

        // delta = softplus(delta_pre + bdt)
        softplus_bias<<<(ROWS * DIc) / 256, 256, 0, stream>>>(delta, bdt);

        // fused scan + gate  -> yb
        mamba_scan<<<(Bc * DIc * DSc) / 256, 256, 0, stream>>>(
            delta, u, dbc, xz, Al, Dl, yb);

        // x = yb @ Wo  (2048 x 2048 x 1024); x is dead, safe to overwrite
        gemm_wmma_tdm<<<dim3(DMc / 64, ROWS / 64), 128, 0, stream>>>(
            yb, DIc, Wo, DMc, x, DMc, DIc);
    }

    // layernorm in place
    layernorm_inplace<<<ROWS, 256, 0, stream>>>(x, ln_g, ln_b);

    // logits = x @ head  (2048 x 1024 x 32000)  [TDM + WMMA]
    gemm_wmma_tdm<<<dim3(Vc / 64, ROWS / 64), 128, 0, stream>>>(
        x, DMc, head, Vc, out, Vc, DMc);
}